// P7_GATSoftmax_41807211659482
// MI455X (gfx1250) — compile-verified
//
#include <hip/hip_runtime.h>
#include <hip/hip_bf16.h>
#include <math.h>

#define BLOCK 256
#define VEC 4

typedef float     v4f __attribute__((ext_vector_type(4)));
typedef int       v4i __attribute__((ext_vector_type(4)));
typedef long long v2l __attribute__((ext_vector_type(2)));

// ---- CDNA5 native no-return FP atomics (device scope, resolved in L2) ----
__device__ __forceinline__ void gmax_f32(float* p, float v) {
    asm volatile("global_atomic_max_num_f32 %0, %1, off scope:SCOPE_DEV"
                 :: "v"(p), "v"(v) : "memory");
}
__device__ __forceinline__ void gadd_f32(float* p, float v) {
    asm volatile("global_atomic_add_f32 %0, %1, off scope:SCOPE_DEV"
                 :: "v"(p), "v"(v) : "memory");
}

// Load 4 segment ids as int32, from either the original int64 stream
// (2x b128 NT loads) or the compacted int32 stream (1x b128 NT load).
__device__ __forceinline__ v4i load4idx(const long long* t, int i0) {
    v2l a = __builtin_nontemporal_load((const v2l*)(t + i0));
    v2l b = __builtin_nontemporal_load((const v2l*)(t + i0 + 2));
    v4i r; r.x = (int)a.x; r.y = (int)a.y; r.z = (int)b.x; r.w = (int)b.y;
    return r;
}
__device__ __forceinline__ v4i load4idx(const int* t, int i0) {
    return __builtin_nontemporal_load((const v4i*)(t + i0));
}

// Pass 0: init node tables (N read from device scalar; grid-stride).
__global__ __launch_bounds__(BLOCK) void k_init(float* __restrict__ emax,
                                                float* __restrict__ esum,
                                                const int* __restrict__ n_ptr) {
    const int n = *n_ptr;
    for (int i = blockIdx.x * BLOCK + threadIdx.x; i < n; i += gridDim.x * BLOCK) {
        emax[i] = -INFINITY;
        esum[i] = 0.0f;
    }
}

// Pass 1: segment max via native f32 atomic max; optionally compact the
// int64 tgt stream to int32 (it must be read here anyway) so passes 2/3
// read half the index bytes.
__global__ __launch_bounds__(BLOCK) void k_segmax(const float* __restrict__ e,
                                                  const long long* __restrict__ tgt,
                                                  float* __restrict__ emax,
                                                  int* __restrict__ t32, int E) {
    const int i0 = (blockIdx.x * BLOCK + threadIdx.x) * VEC;
    if (i0 + VEC <= E) {
        v4f ev = __builtin_nontemporal_load((const v4f*)(e + i0));
        v4i t  = load4idx(tgt, i0);
        if (t32) __builtin_nontemporal_store(t, (v4i*)(t32 + i0));
        gmax_f32(emax + t.x, ev.x);
        gmax_f32(emax + t.y, ev.y);
        gmax_f32(emax + t.z, ev.z);
        gmax_f32(emax + t.w, ev.w);
    } else {
        for (int i = i0; i < E; ++i) {
            int ti = (int)tgt[i];
            if (t32) t32[i] = ti;
            gmax_f32(emax + ti, e[i]);
        }
    }
}

// Pass 2: exp(e - max[tgt]) accumulated via native f32 atomic add.
// emax gathers are regular (RT) loads -> L2 hits (2 MB table, 192 MB L2).
template <typename IT>
__global__ __launch_bounds__(BLOCK) void k_expsum(const float* __restrict__ e,
                                                  const IT* __restrict__ tgt,
                                                  const float* __restrict__ emax,
                                                  float* __restrict__ esum, int E) {
    const int i0 = (blockIdx.x * BLOCK + threadIdx.x) * VEC;
    if (i0 + VEC <= E) {
        v4f ev = __builtin_nontemporal_load((const v4f*)(e + i0));
        v4i t  = load4idx(tgt, i0);
        gadd_f32(esum + t.x, __expf(ev.x - emax[t.x]));
        gadd_f32(esum + t.y, __expf(ev.y - emax[t.y]));
        gadd_f32(esum + t.z, __expf(ev.z - emax[t.z]));
        gadd_f32(esum + t.w, __expf(ev.w - emax[t.w]));
    } else {
        for (int i = i0; i < E; ++i) {
            int t = (int)tgt[i];
            gadd_f32(esum + t, __expf(e[i] - emax[t]));
        }
    }
}

// Pass 3: alpha = exp(e - max) / (sum + 1e-16). exp recomputed (VALU is free
// at 23 TB/s; saves a 128 MB scratch round-trip). Streaming NT store.
template <typename IT>
__global__ __launch_bounds__(BLOCK) void k_norm(const float* __restrict__ e,
                                                const IT* __restrict__ tgt,
                                                const float* __restrict__ emax,
                                                const float* __restrict__ esum,
                                                float* __restrict__ alpha, int E) {
    const int i0 = (blockIdx.x * BLOCK + threadIdx.x) * VEC;
    if (i0 + VEC <= E) {
        v4f ev = __builtin_nontemporal_load((const v4f*)(e + i0));
        v4i t  = load4idx(tgt, i0);
        v4f r;
        r.x = __expf(ev.x - emax[t.x]) / (esum[t.x] + 1e-16f);
        r.y = __expf(ev.y - emax[t.y]) / (esum[t.y] + 1e-16f);
        r.z = __expf(ev.z - emax[t.z]) / (esum[t.z] + 1e-16f);
        r.w = __expf(ev.w - emax[t.w]) / (esum[t.w] + 1e-16f);
        __builtin_nontemporal_store(r, (v4f*)(alpha + i0));
    } else {
        for (int i = i0; i < E; ++i) {
            int t = (int)tgt[i];
            alpha[i] = __expf(e[i] - emax[t]) / (esum[t] + 1e-16f);
        }
    }
}

extern "C" void kernel_launch(void* const* d_in, const int* in_sizes, int n_in,
                              void* d_out, int out_size, void* d_ws, size_t ws_size,
                              hipStream_t stream) {
    const float*     e    = (const float*)d_in[0];
    const long long* eidx = (const long long*)d_in[1];   // int64 [2, E] flat
    const int*       nptr = (const int*)d_in[2];         // num_nodes (LE-safe read)

    const int E = in_sizes[0];                 // 32,000,000
    const long long* tgt = eidx + E;           // row 1 = target node ids

    // Workspace: emax[N] at 0, esum[N] at +2 MB (524288 slots >= N=500000),
    // optional int32 index copy at +4 MB.
    float* emax  = (float*)d_ws;
    float* esum  = (float*)d_ws + (1 << 19);
    int*   t32   = (int*)((char*)d_ws + (4u << 20));
    float* alpha = (float*)d_out;

    const bool compact =
        ws_size >= (size_t)(4u << 20) + (size_t)E * sizeof(int);

    k_init<<<1024, BLOCK, 0, stream>>>(emax, esum, nptr);

    const int blocks = (E + BLOCK * VEC - 1) / (BLOCK * VEC);  // 31250
    if (compact) {
        k_segmax<<<blocks, BLOCK, 0, stream>>>(e, tgt, emax, t32, E);
        k_expsum<int><<<blocks, BLOCK, 0, stream>>>(e, t32, emax, esum, E);
        k_norm<int>  <<<blocks, BLOCK, 0, stream>>>(e, t32, emax, esum, alpha, E);
    } else {
        k_segmax<<<blocks, BLOCK, 0, stream>>>(e, tgt, emax, (int*)nullptr, E);
        k_expsum<long long><<<blocks, BLOCK, 0, stream>>>(e, tgt, emax, esum, E);
        k_norm<long long>  <<<blocks, BLOCK, 0, stream>>>(e, tgt, emax, esum, alpha, E);
    }
}